// gcn_layer_68573447848481
// MI455X (gfx1250) — compile-verified
//
#include <hip/hip_runtime.h>
#include <stdint.h>

#define GDIM 128
#define TPB 256
#define CHUNK 256   // edges staged per block

// ---------------------------------------------------------------- zero init
__global__ void gcn_zero_kernel(float4* __restrict__ out, int n4) {
  int i = blockIdx.x * blockDim.x + threadIdx.x;
  if (i < n4) out[i] = make_float4(0.f, 0.f, 0.f, 0.f);
}

// ---------------------------------------------------------------- edge scatter
// One wave (32 lanes) processes one edge at a time: lane l handles features
// [4l, 4l+4). Edge triples are staged into LDS with CDNA5 async global->LDS
// loads; each wave consumes exactly the 32 slots its own lanes issued, so a
// wave-local s_wait_asynccnt is sufficient (no workgroup barrier needed).
__global__ __launch_bounds__(TPB) void gcn_edge_scatter_kernel(
    const float* __restrict__ x, const float* __restrict__ W,
    const int* __restrict__ erow, const int* __restrict__ ecol,
    const float* __restrict__ eval_, float* __restrict__ out, int nEdges) {
  __shared__ int   s_row[CHUNK];
  __shared__ int   s_col[CHUNK];
  __shared__ float s_val[CHUNK];

  const int t    = threadIdx.x;
  const int lane = t & 31;
  const int wave = t >> 5;

  // per-lane slice of the [1,128] feature scale, loaded once
  const float4 wv = ((const float4*)W)[lane];

  const int base = blockIdx.x * CHUNK;
  const int e    = base + t;

  if (e < nEdges) {
#if defined(__gfx1250__)
    // LDS byte offsets: generic LDS pointers carry the group-segment offset
    // in their low 32 bits (aperture in the high bits).
    uint32_t l_row = (uint32_t)(uintptr_t)&s_row[t];
    uint32_t l_col = (uint32_t)(uintptr_t)&s_col[t];
    uint32_t l_val = (uint32_t)(uintptr_t)&s_val[t];
    uint64_t g_row = (uint64_t)(uintptr_t)(erow  + e);
    uint64_t g_col = (uint64_t)(uintptr_t)(ecol  + e);
    uint64_t g_val = (uint64_t)(uintptr_t)(eval_ + e);
    asm volatile(
        "global_load_async_to_lds_b32 %0, %3, off\n\t"
        "global_load_async_to_lds_b32 %1, %4, off\n\t"
        "global_load_async_to_lds_b32 %2, %5, off"
        :
        : "v"(l_row), "v"(l_col), "v"(l_val),
          "v"(g_row), "v"(g_col), "v"(g_val)
        : "memory");
#else
    s_row[t] = erow[e];
    s_col[t] = ecol[e];
    s_val[t] = eval_[e];
#endif
  }
#if defined(__gfx1250__)
  asm volatile("s_wait_asynccnt 0" ::: "memory");
#endif

  // Each wave consumes its own 32 staged edges.
  const int jbase = wave * 32;
  int c = s_col[jbase];  // software-pipelined column index (for prefetch)
  for (int j = 0; j < 32; ++j) {
    const int idx = jbase + j;
    const int eg  = base + idx;
    if (eg >= nEdges) break;

    const int   r = s_row[idx];
    const float v = s_val[idx];
    const int  cn = (j + 1 < 32) ? s_col[idx + 1] : c;

    // coalesced 512B row gather: 32 lanes x float4
    const float4 xv = ((const float4*)x)[(size_t)c * 32 + lane];

    // prefetch next edge's source row while we do the atomics
    if (j + 1 < 32)
      __builtin_prefetch((const float4*)x + (size_t)cn * 32 + lane, 0, 3);

    float4 m;
    m.x = xv.x * wv.x * v;
    m.y = xv.y * wv.y * v;
    m.z = xv.z * wv.z * v;
    m.w = xv.w * wv.w * v;

    float* o = out + (size_t)r * GDIM + lane * 4;
    unsafeAtomicAdd(o + 0, m.x);   // -> global_atomic_add_f32 (no return)
    unsafeAtomicAdd(o + 1, m.y);
    unsafeAtomicAdd(o + 2, m.z);
    unsafeAtomicAdd(o + 3, m.w);

    c = cn;
  }
}

// ---------------------------------------------------------------- relu
__global__ void gcn_relu_kernel(float4* __restrict__ out, int n4) {
  int i = blockIdx.x * blockDim.x + threadIdx.x;
  if (i < n4) {
    float4 v = out[i];
    v.x = fmaxf(v.x, 0.f);
    v.y = fmaxf(v.y, 0.f);
    v.z = fmaxf(v.z, 0.f);
    v.w = fmaxf(v.w, 0.f);
    out[i] = v;
  }
}

// ---------------------------------------------------------------- launch
extern "C" void kernel_launch(void* const* d_in, const int* in_sizes, int n_in,
                              void* d_out, int out_size, void* d_ws, size_t ws_size,
                              hipStream_t stream) {
  const float* x    = (const float*)d_in[0];   // [N,128] f32
  const float* W    = (const float*)d_in[1];   // [1,128] f32
  const int*   erow = (const int*)  d_in[2];   // [E] i32
  const int*   ecol = (const int*)  d_in[3];   // [E] i32
  const float* ev   = (const float*)d_in[4];   // [E] f32
  float*       out  = (float*)d_out;           // [N,128] f32

  const int nEdges = in_sizes[2];
  const int n4     = out_size / 4;

  const int zb = (n4 + TPB - 1) / TPB;
  gcn_zero_kernel<<<zb, TPB, 0, stream>>>((float4*)out, n4);

  const int nb = (nEdges + CHUNK - 1) / CHUNK;   // 6250 blocks for E=1.6M
  gcn_edge_scatter_kernel<<<nb, TPB, 0, stream>>>(x, W, erow, ecol, ev, out, nEdges);

  gcn_relu_kernel<<<zb, TPB, 0, stream>>>((float4*)out, n4);
}